// DCELoss_56461640073337
// MI455X (gfx1250) — compile-verified
//
#include <hip/hip_runtime.h>
#include <stdint.h>

typedef __attribute__((ext_vector_type(2))) float v2f;
typedef __attribute__((ext_vector_type(8))) float v8f;
typedef int async_i2 __attribute__((__vector_size__(2 * sizeof(int))));  // b64 payload type

#define BINS        200
#define HW          98304                    // 256*384 pixels per image
#define LDS_STRIDE  202                      // padded W row stride (floats)
#define THREADS     256
#define PIX_PER_BLK 512                      // 2 pixels per thread
#define BLK_PER_IMG (HW / PIX_PER_BLK)       // 192
#define NBLK        (4 * BLK_PER_IMG)        // 768 blocks total
#define INTERVAL_F  0.00951544993495971795f  // float32(log10(80)/200)
#define EPS         1e-6f

// ---------- gfx1250 async global->LDS helpers ----------
// Builtin signature (from clang diagnostic): arg0 = AS1 pointer to int __vector(2),
// arg1 = AS3 pointer, then two constant ints (offset, cpol).
#define AS1V2(p) ((__attribute__((address_space(1))) async_i2*)(uintptr_t)(p))
#define AS3V2(p) ((__attribute__((address_space(3))) async_i2*)(uintptr_t)(p))

#if defined(__HIP_DEVICE_COMPILE__) && __has_builtin(__builtin_amdgcn_global_load_async_to_lds_b64)
#define ASYNC_B64(g, l) \
  __builtin_amdgcn_global_load_async_to_lds_b64(AS1V2(g), AS3V2(l), 0, 0)
#else
#define ASYNC_B64(g, l)                                                       \
  asm volatile("global_load_async_to_lds_b64 %0, %1, off" ::                  \
               "v"((uint32_t)(uintptr_t)(l)), "v"((uint64_t)(uintptr_t)(g)) : \
               "memory")
#endif

#if defined(__HIP_DEVICE_COMPILE__) && __has_builtin(__builtin_amdgcn_s_wait_asynccnt)
#define WAIT_ASYNC0() __builtin_amdgcn_s_wait_asynccnt(0)
#else
#define WAIT_ASYNC0() asm volatile("s_wait_asynccnt 0x0" ::: "memory")
#endif

#if defined(__HIP_DEVICE_COMPILE__) && __has_builtin(__builtin_amdgcn_wmma_f32_16x16x4_f32)
#define HAVE_WMMA_F32 1
#endif

// Matches reference depth_to_bins_in_log ordering of clamps; invalid -> 201.
__device__ __forceinline__ int depth_bin(float d, bool msk) {
  if (!msk) return BINS + 1;
  if (d <= 1.0f) return 0;
  if (d >= 80.0f) return BINS - 1;
  int b = (int)(log10f(d) / INTERVAL_F);  // trunc toward zero, d>1 => b>=0
  if (b > BINS - 1) b = BINS - 1;
  return b;
}

__global__ __launch_bounds__(THREADS)
void dce_main(const float* __restrict__ target,
              const unsigned char* __restrict__ mask,
              const float* __restrict__ logits,
              const float* __restrict__ binsw,
              float* __restrict__ part) {
  __shared__ float lds_w[BINS * LDS_STRIDE];
  __shared__ float redl[THREADS / 32];
  __shared__ float redc[THREADS / 32];

  const int tid = threadIdx.x;

  // ---- stage bins_weight (200x200 f32) into LDS with async b64 copies ----
  // 100 8-byte chunks per row; padded row stride keeps 8B alignment.
  const int chunks = BINS * (BINS / 2);  // 20000
  for (int i = tid; i < chunks; i += THREADS) {
    int g = i / 100;
    int k = i - g * 100;
    const float* src = binsw + g * BINS + k * 2;
    float* dst = lds_w + g * LDS_STRIDE + k * 2;
    ASYNC_B64(src, dst);
  }
  WAIT_ASYNC0();
  __syncthreads();

  // ---- 2 pixels per thread, block fully inside one image ----
  const int b    = blockIdx.x / BLK_PER_IMG;
  const int wib  = blockIdx.x - b * BLK_PER_IMG;
  const int pix0 = wib * PIX_PER_BLK + tid;
  const size_t t0 = (size_t)b * HW + pix0;
  const size_t t1 = t0 + THREADS;

  const bool mk0 = mask[t0] != 0;
  const bool mk1 = mask[t1] != 0;
  const int bin0 = depth_bin(target[t0], mk0);
  const int bin1 = depth_bin(target[t1], mk1);
  const bool v0 = bin0 < BINS;
  const bool v1 = bin1 < BINS;
  const int g0 = v0 ? bin0 : 0;
  const int g1 = v1 ? bin1 : 0;

  const float* xp0 = logits + (size_t)b * BINS * HW + pix0;
  const float* xp1 = xp0 + THREADS;
  const float* wr0 = lds_w + g0 * LDS_STRIDE;
  const float* wr1 = lds_w + g1 * LDS_STRIDE;

  // online logsumexp + raw-logit dot + weight-row sum, single pass over C
  float m0 = -3.0e38f, s0 = 0.f, dot0 = 0.f, ws0 = 0.f;
  float m1 = -3.0e38f, s1 = 0.f, dot1 = 0.f, ws1 = 0.f;
#pragma unroll 4
  for (int c = 0; c < BINS; ++c) {
    const float x0 = xp0[(size_t)c * HW];
    const float x1 = xp1[(size_t)c * HW];
    const float w0 = wr0[c];
    const float w1 = wr1[c];
    dot0 = fmaf(w0, x0, dot0); ws0 += w0;
    dot1 = fmaf(w1, x1, dot1); ws1 += w1;
    const float nm0 = fmaxf(m0, x0);
    s0 = fmaf(s0, __expf(m0 - nm0), __expf(x0 - nm0)); m0 = nm0;
    const float nm1 = fmaxf(m1, x1);
    s1 = fmaf(s1, __expf(m1 - nm1), __expf(x1 - nm1)); m1 = nm1;
  }

  // loss_p = wsum*(max + log s) - dot ; zero for invalid pixels
  float lp = 0.f, cn = 0.f;
  if (v0) { lp += ws0 * (m0 + __logf(s0)) - dot0; cn += 1.f; }
  if (v1) { lp += ws1 * (m1 + __logf(s1)) - dot1; cn += 1.f; }

  // ---- wave32 shuffle reduction, then cross-wave via LDS ----
  for (int off = 16; off > 0; off >>= 1) {
    lp += __shfl_xor(lp, off, 32);
    cn += __shfl_xor(cn, off, 32);
  }
  const int wv = tid >> 5;
  if ((tid & 31) == 0) { redl[wv] = lp; redc[wv] = cn; }
  __syncthreads();
  if (tid == 0) {
    float tl = 0.f, tc = 0.f;
    for (int w = 0; w < THREADS / 32; ++w) { tl += redl[w]; tc += redc[w]; }
    part[blockIdx.x] = tl;
    part[NBLK + blockIdx.x] = tc;
  }
}

// Reduce 768 loss partials + 768 mask-count partials with f32 WMMA:
// D = A(16x4 partials) x B(4x16 ones) + C accumulates per-row sums in f32.
__global__ __launch_bounds__(32)
void dce_finalize(const float* __restrict__ part, float* __restrict__ out) {
  const int lane = threadIdx.x;
  float tL, tC;
#ifdef HAVE_WMMA_F32
  v8f accL = {0.f, 0.f, 0.f, 0.f, 0.f, 0.f, 0.f, 0.f};
  v8f accC = {0.f, 0.f, 0.f, 0.f, 0.f, 0.f, 0.f, 0.f};
  const v2f ones = {1.f, 1.f};
#pragma unroll
  for (int ch = 0; ch < NBLK / 64; ++ch) {  // 12 chunks of 64
    const int base = ch * 64;
    v2f aL, aC;
    aL.x = part[base + lane];
    aL.y = part[base + 32 + lane];
    aC.x = part[NBLK + base + lane];
    aC.y = part[NBLK + base + 32 + lane];
    accL = __builtin_amdgcn_wmma_f32_16x16x4_f32(false, aL, false, ones,
                                                 (short)0, accL, false, false);
    accC = __builtin_amdgcn_wmma_f32_16x16x4_f32(false, aC, false, ones,
                                                 (short)0, accC, false, false);
  }
  // lanes 0-15 hold rows 0-7 (one per VGPR), lanes 16-31 hold rows 8-15
  tL = 0.f; tC = 0.f;
#pragma unroll
  for (int r = 0; r < 8; ++r) { tL += accL[r]; tC += accC[r]; }
  tL += __shfl_xor(tL, 16, 32);
  tC += __shfl_xor(tC, 16, 32);
#else
  tL = 0.f; tC = 0.f;
  for (int i = lane; i < NBLK; i += 32) { tL += part[i]; tC += part[NBLK + i]; }
  for (int off = 16; off > 0; off >>= 1) {
    tL += __shfl_xor(tL, off, 32);
    tC += __shfl_xor(tC, off, 32);
  }
#endif
  if (lane == 0) out[0] = tL / (tC + EPS);  // LOSS_WEIGHT == 1.0
}

extern "C" void kernel_launch(void* const* d_in, const int* in_sizes, int n_in,
                              void* d_out, int out_size, void* d_ws, size_t ws_size,
                              hipStream_t stream) {
  const float*         target = (const float*)d_in[0];
  const unsigned char* mask   = (const unsigned char*)d_in[1];  // jnp bool -> 1 byte
  const float*         logits = (const float*)d_in[2];
  const float*         binsw  = (const float*)d_in[3];
  float* part = (float*)d_ws;  // [0..NBLK): loss partials, [NBLK..2*NBLK): counts

  dce_main<<<NBLK, THREADS, 0, stream>>>(target, mask, logits, binsw, part);
  dce_finalize<<<1, 32, 0, stream>>>(part, (float*)d_out);
}